// E2VLayer_17669495456077
// MI455X (gfx1250) — compile-verified
//
#include <hip/hip_runtime.h>
#include <math.h>

#define N_NODES 100000
#define N_EDGES 1600000
#define DE 48
#define KDIM 144   /* 3*DE */
#define DX 128

typedef float v2f __attribute__((ext_vector_type(2)));
typedef float v8f __attribute__((ext_vector_type(8)));

// ---------------- utility kernels ----------------
__global__ void zero_i32(int* __restrict__ p, int n) {
  int i = blockIdx.x * blockDim.x + threadIdx.x;
  if (i < n) p[i] = 0;
}

__global__ void copy_i32(const int* __restrict__ s, int* __restrict__ d, int n) {
  int i = blockIdx.x * blockDim.x + threadIdx.x;
  if (i < n) d[i] = s[i];
}

// ---------------- CSR build ----------------
__global__ void count_edges(const int* __restrict__ dst, int* __restrict__ cnt, int n) {
  int i = blockIdx.x * blockDim.x + threadIdx.x;
  if (i < n) atomicAdd(&cnt[dst[i]], 1);
}

// single-workgroup chunked exclusive scan (100K ints -> 98 chunks of 1024)
__global__ void scan_counts(const int* __restrict__ cnt, int* __restrict__ offs, int n) {
  __shared__ int sh[1024];
  __shared__ int carry;
  const int tid = threadIdx.x;
  if (tid == 0) carry = 0;
  __syncthreads();
  for (int base = 0; base < n; base += 1024) {
    int i = base + tid;
    int v = (i < n) ? cnt[i] : 0;
    sh[tid] = v;
    __syncthreads();
    for (int off = 1; off < 1024; off <<= 1) {
      int t = (tid >= off) ? sh[tid - off] : 0;
      __syncthreads();
      sh[tid] += t;
      __syncthreads();
    }
    if (i < n) offs[i] = carry + sh[tid] - v;   // exclusive
    __syncthreads();
    if (tid == 1023) carry += sh[1023];
    __syncthreads();
  }
  if (tid == 0) offs[n] = carry;
}

__global__ void fill_csr(const int* __restrict__ dst, int* __restrict__ cursor,
                         int* __restrict__ csr, int n) {
  int i = blockIdx.x * blockDim.x + threadIdx.x;
  if (i < n) {
    int p = atomicAdd(&cursor[dst[i]], 1);
    csr[p] = i;
  }
}

// ---------------- per-node gather reduce (one wave32 per node) ----------------
// lane handles column `lane`; lanes 0..15 additionally handle column 32+lane.
// Mean accumulated in int64 fixed point (scale 2^20) -> bit-deterministic
// regardless of CSR fill order; min/max are order-independent.
__global__ void gather_reduce(const float* __restrict__ fe, const int* __restrict__ csr,
                              const int* __restrict__ offs, float* __restrict__ z) {
  const int lane = threadIdx.x & 31;
  const int node = (blockIdx.x * blockDim.x + threadIdx.x) >> 5;
  if (node >= N_NODES) return;
  const int beg = offs[node];
  const int end = offs[node + 1];
  const int deg = end - beg;
  long long s0 = 0, s1 = 0;
  float mn0 = INFINITY, mx0 = -INFINITY;
  float mn1 = INFINITY, mx1 = -INFINITY;
  for (int j = beg; j < end; ++j) {
    const float* row = fe + (size_t)csr[j] * DE;   // 192B row, coalesced per wave
    float v0 = row[lane];
    s0 += (long long)__float2int_rn(v0 * 1048576.0f);
    mn0 = fminf(mn0, v0);
    mx0 = fmaxf(mx0, v0);
    if (lane < 16) {
      float v1 = row[32 + lane];
      s1 += (long long)__float2int_rn(v1 * 1048576.0f);
      mn1 = fminf(mn1, v1);
      mx1 = fmaxf(mx1, v1);
    }
  }
  float* zr = z + (size_t)node * KDIM;
  if (deg > 0) {
    double inv = 1.0 / (1048576.0 * (double)deg);
    zr[lane]      = (float)((double)s0 * inv);
    zr[48 + lane] = mn0;
    zr[96 + lane] = mx0;
    if (lane < 16) {
      zr[32 + lane]  = (float)((double)s1 * inv);
      zr[80 + lane]  = mn1;
      zr[128 + lane] = mx1;
    }
  } else {
    zr[lane] = 0.0f; zr[48 + lane] = 0.0f; zr[96 + lane] = 0.0f;
    if (lane < 16) { zr[32 + lane] = 0.0f; zr[80 + lane] = 0.0f; zr[128 + lane] = 0.0f; }
  }
}

// ---------------- WMMA projection: out[N,128] = z[N,144] @ W^T + b ----------------
// One wave per 16-row stripe, all 8 N-tiles held in registers (8 x v8f acc).
// v_wmma_f32_16x16x4_f32 fragment layout (ISA 7.12.2, 32-bit A 16x4):
//   A: lane -> row M = lane&15, K pair = k0 + 2*(lane>>4) (contiguous float2 in z)
//   B: lane -> col N = lane&15, same K pair  (contiguous float2 in W row-major [128,144])
//   C/D: VGPR i -> row M = i + 8*(lane>>4), col N = lane&15
// Inner loop: batch-load A + all 8 B fragments into live registers, then issue
// 8 WMMAs; full unroll over the 36 K-steps lets the scheduler hoist the next
// step's loads above the current step's WMMAs (register-renamed double buffer)
// instead of the load/s_wait/wmma lockstep of the previous version.
__global__ void node_gemm(const float* __restrict__ z, const float* __restrict__ W,
                          const float* __restrict__ b, float* __restrict__ out) {
  const int lane = threadIdx.x & 31;
  const int gw = (blockIdx.x * blockDim.x + threadIdx.x) >> 5;
  const int mtiles = N_NODES / 16;   // 6250 exactly
  if (gw >= mtiles) return;
  const int g = lane >> 4;
  const int c = lane & 15;
  const int row0 = gw * 16;

  v8f acc[8];
#pragma unroll
  for (int t = 0; t < 8; ++t) acc[t] = (v8f){0.f, 0.f, 0.f, 0.f, 0.f, 0.f, 0.f, 0.f};

  const float* arow = z + (size_t)(row0 + c) * KDIM + 2 * g;  // A row for this lane
  const float* wcol = W + (size_t)c * KDIM + 2 * g;           // B base for this lane

#pragma unroll
  for (int s = 0; s < KDIM / 4; ++s) {
    const int k0 = s * 4;
    // batch all 9 loads so the wave waits once, not once per WMMA
    v2f a = *(const v2f*)(arow + k0);
    v2f bb[8];
#pragma unroll
    for (int t = 0; t < 8; ++t)
      bb[t] = *(const v2f*)(wcol + (size_t)t * 16 * KDIM + k0);
#pragma unroll
    for (int t = 0; t < 8; ++t)
      acc[t] = __builtin_amdgcn_wmma_f32_16x16x4_f32(
          /*neg_a=*/false, a, /*neg_b=*/false, bb[t],
          /*c_mod=*/(short)0, acc[t], /*reuse_a=*/false, /*reuse_b=*/false);
  }

#pragma unroll
  for (int t = 0; t < 8; ++t) {
    float bias = b[t * 16 + c];
#pragma unroll
    for (int i = 0; i < 8; ++i) {
      int row = row0 + i + 8 * g;
      out[(size_t)row * DX + t * 16 + c] = acc[t][i] + bias;
    }
  }
}

// ---------------- launch ----------------
extern "C" void kernel_launch(void* const* d_in, const int* in_sizes, int n_in,
                              void* d_out, int out_size, void* d_ws, size_t ws_size,
                              hipStream_t stream) {
  const float* fe = (const float*)d_in[0];
  const int* dst  = (const int*)d_in[1];
  const float* W  = (const float*)d_in[2];
  const float* b  = (const float*)d_in[3];
  float* out = (float*)d_out;

  char* ws = (char*)d_ws;
  int*   cnt    = (int*)(ws + 0);          // 400000 B
  int*   offs   = (int*)(ws + 400000);     // 400004 B
  int*   cursor = (int*)(ws + 800064);     // 400000 B
  int*   csr    = (int*)(ws + 1200064);    // 6400000 B
  float* z      = (float*)(ws + 7600128);  // 57600000 B  (~65 MB total)

  zero_i32   <<<(N_NODES + 255) / 256, 256, 0, stream>>>(cnt, N_NODES);
  count_edges<<<(N_EDGES + 255) / 256, 256, 0, stream>>>(dst, cnt, N_EDGES);
  scan_counts<<<1, 1024, 0, stream>>>(cnt, offs, N_NODES);
  copy_i32   <<<(N_NODES + 255) / 256, 256, 0, stream>>>(offs, cursor, N_NODES);
  fill_csr   <<<(N_EDGES + 255) / 256, 256, 0, stream>>>(dst, cursor, csr, N_EDGES);
  gather_reduce<<<(N_NODES * 32 + 255) / 256, 256, 0, stream>>>(fe, csr, offs, z);
  node_gemm  <<<(N_NODES / 16 + 3) / 4, 128, 0, stream>>>(z, W, b, out);
}